// ASQRT_autograd_mc_1915555414730
// MI455X (gfx1250) — compile-verified
//
#include <hip/hip_runtime.h>
#include <math.h>

typedef float v2f __attribute__((ext_vector_type(2)));
typedef float v8f __attribute__((ext_vector_type(8)));

#define NMAT     128
#define LDA      132            // padded row stride (floats): 132 % 64 = 4 banks -> conflict-free A loads
#define NN       (NMAT * NMAT)
#define NUM_ITER 5
#define THREADS  256            // 8 wave32

// One wave computes a 4(m) x 2(n) block of 16x16 output tiles of D = SA * SB
// using V_WMMA_F32_16X16X4_F32, accumulating over K=128 in chunks of 4.
// Fragment layouts per CDNA5 ISA 7.12.2 (f32, wave32):
//   A 16x4 : lane holds row (lane&15); VGPR v holds K = k0 + 2*(lane>>4) + v
//   B 4x16 : lane holds col (lane&15); VGPR v holds K = k0 + 2*(lane>>4) + v (row of B)
//   C/D    : VGPR j holds row j + 8*(lane>>4), col (lane&15)
__device__ __forceinline__ void wmma_mm128(const float* __restrict__ SA,
                                           const float* __restrict__ SB,
                                           int wrow, int wcol, int lane,
                                           v8f acc[4][2]) {
  const int r  = lane & 15;
  const int h2 = (lane >> 4) << 1;                 // 0 or 2
  const float* Abase = SA + (wrow + r) * LDA + h2; // + k
  const float* Bbase = SB + h2 * LDA + wcol + r;   // + k*LDA (+ nt*16)
#pragma unroll 4
  for (int k = 0; k < NMAT; k += 4) {
    const float* ap = Abase + k;
    const float* bp = Bbase + k * LDA;
    v2f a0, a1, a2, a3, b0, b1;
    a0.x = ap[0];              a0.y = ap[1];              // rows wrow+0..15
    a1.x = ap[16 * LDA];       a1.y = ap[16 * LDA + 1];   // rows wrow+16..31
    a2.x = ap[32 * LDA];       a2.y = ap[32 * LDA + 1];   // rows wrow+32..47
    a3.x = ap[48 * LDA];       a3.y = ap[48 * LDA + 1];   // rows wrow+48..63
    b0.x = bp[0];              b0.y = bp[LDA];            // cols wcol+0..15
    b1.x = bp[16];             b1.y = bp[LDA + 16];       // cols wcol+16..31
    acc[0][0] = __builtin_amdgcn_wmma_f32_16x16x4_f32(false, a0, false, b0, (short)0, acc[0][0], false, false);
    acc[0][1] = __builtin_amdgcn_wmma_f32_16x16x4_f32(false, a0, false, b1, (short)0, acc[0][1], false, false);
    acc[1][0] = __builtin_amdgcn_wmma_f32_16x16x4_f32(false, a1, false, b0, (short)0, acc[1][0], false, false);
    acc[1][1] = __builtin_amdgcn_wmma_f32_16x16x4_f32(false, a1, false, b1, (short)0, acc[1][1], false, false);
    acc[2][0] = __builtin_amdgcn_wmma_f32_16x16x4_f32(false, a2, false, b0, (short)0, acc[2][0], false, false);
    acc[2][1] = __builtin_amdgcn_wmma_f32_16x16x4_f32(false, a2, false, b1, (short)0, acc[2][1], false, false);
    acc[3][0] = __builtin_amdgcn_wmma_f32_16x16x4_f32(false, a3, false, b0, (short)0, acc[3][0], false, false);
    acc[3][1] = __builtin_amdgcn_wmma_f32_16x16x4_f32(false, a3, false, b1, (short)0, acc[3][1], false, false);
  }
}

__device__ __forceinline__ void zero_acc(v8f acc[4][2]) {
#pragma unroll
  for (int mt = 0; mt < 4; ++mt)
#pragma unroll
    for (int nt = 0; nt < 2; ++nt)
      acc[mt][nt] = (v8f){0.f, 0.f, 0.f, 0.f, 0.f, 0.f, 0.f, 0.f};
}

extern "C" __global__ void __launch_bounds__(THREADS)
ns_msqrt_wmma(const float* __restrict__ A, float* __restrict__ out) {
  extern __shared__ float smem[];
  float* sY  = smem;
  float* sZ  = sY + NMAT * LDA;
  float* sT  = sZ + NMAT * LDA;
  float* red = sT + NMAT * LDA;   // 8 floats for cross-wave reduction

  const int tid  = threadIdx.x;
  const int lane = tid & 31;
  const int wave = tid >> 5;
  const int wrow = (wave >> 2) * 64;  // 4 m-tiles starting here
  const int wcol = (wave & 3) * 32;   // 2 n-tiles starting here

  const float* Ab = A   + (size_t)blockIdx.x * NN;
  float*       Ob = out + (size_t)blockIdx.x * NN;

  // ---- Frobenius norm: sum(A*A) ----
  float ss = 0.0f;
  for (int i = tid; i < NN; i += THREADS) { float v = Ab[i]; ss += v * v; }
#pragma unroll
  for (int off = 16; off > 0; off >>= 1) ss += __shfl_down(ss, off, 32);
  if (lane == 0) red[wave] = ss;
  __syncthreads();
  float sumsq = 0.0f;
#pragma unroll
  for (int w = 0; w < 8; ++w) sumsq += red[w];
  const float normA = sqrtf(sumsq);
  const float rinv  = 1.0f / normA;

  // ---- Init: Y = A / ||A||_F, Z = I ----
  for (int i = tid; i < NN; i += THREADS) {
    const int rr = i >> 7, cc = i & (NMAT - 1);
    sY[rr * LDA + cc] = Ab[i] * rinv;
    sZ[rr * LDA + cc] = (rr == cc) ? 1.0f : 0.0f;
  }
  __syncthreads();

  const int r  = lane & 15;
  const int h8 = (lane >> 4) << 3;

  // ---- Newton-Schulz iterations ----
  for (int it = 0; it < NUM_ITER; ++it) {
    // Phase 1: P = Z @ Y  ->  T = 1.5*I - 0.5*P
    v8f acc[4][2];
    zero_acc(acc);
    wmma_mm128(sZ, sY, wrow, wcol, lane, acc);
#pragma unroll
    for (int mt = 0; mt < 4; ++mt)
#pragma unroll
      for (int nt = 0; nt < 2; ++nt)
#pragma unroll
        for (int j = 0; j < 8; ++j) {
          const int rr = wrow + mt * 16 + h8 + j;
          const int cc = wcol + nt * 16 + r;
          sT[rr * LDA + cc] = ((rr == cc) ? 1.5f : 0.0f) - 0.5f * acc[mt][nt][j];
        }
    __syncthreads();

    // Phase 2: Ynew = Y @ T ; Znew = T @ Z  (both held in registers)
    v8f accY[4][2], accZ[4][2];
    zero_acc(accY);
    zero_acc(accZ);
    wmma_mm128(sY, sT, wrow, wcol, lane, accY);
    wmma_mm128(sT, sZ, wrow, wcol, lane, accZ);
    __syncthreads();   // all reads of sY/sZ complete before overwrite
#pragma unroll
    for (int mt = 0; mt < 4; ++mt)
#pragma unroll
      for (int nt = 0; nt < 2; ++nt)
#pragma unroll
        for (int j = 0; j < 8; ++j) {
          const int rr = wrow + mt * 16 + h8 + j;
          const int cc = wcol + nt * 16 + r;
          sY[rr * LDA + cc] = accY[mt][nt][j];
          sZ[rr * LDA + cc] = accZ[mt][nt][j];
        }
    __syncthreads();
  }

  // ---- Output: sA = Y * sqrt(||A||_F) ----
  const float sc = sqrtf(normA);
  for (int i = tid; i < NN; i += THREADS) {
    const int rr = i >> 7, cc = i & (NMAT - 1);
    Ob[i] = sY[rr * LDA + cc] * sc;
  }
}

extern "C" void kernel_launch(void* const* d_in, const int* in_sizes, int n_in,
                              void* d_out, int out_size, void* d_ws, size_t ws_size,
                              hipStream_t stream) {
  (void)n_in; (void)out_size; (void)d_ws; (void)ws_size;
  const float* A = (const float*)d_in[0];
  float* out = (float*)d_out;
  const int num_mats = in_sizes[0] / NN;           // 32*32 = 1024
  const size_t shmem = (size_t)(3 * NMAT * LDA + 8) * sizeof(float);  // ~198 KB < 320 KB WGP LDS
  ns_msqrt_wmma<<<num_mats, THREADS, shmem, stream>>>(A, out);
}